// HER_2276332666917
// MI455X (gfx1250) — compile-verified
//
#include <hip/hip_runtime.h>
#include <cstdint>

typedef __attribute__((ext_vector_type(2))) float v2f;
typedef __attribute__((ext_vector_type(8))) float v8f;

#define DDIM 8192
#define RDIM 4096
#define KCHUNKS 16
#define KC (DDIM / KCHUNKS)   // 512 K-rows per wave
#define ALPHA_L 0.075f
#define GAMMA 10.0f

// ---------------------------------------------------------------------------
// Threefry2x32, key = (0, 42)  (jax.random.key(42)), 20 rounds — exact JAX RNG
// ---------------------------------------------------------------------------
__device__ __forceinline__ void threefry2x32(uint32_t c0, uint32_t c1,
                                             uint32_t& o0, uint32_t& o1) {
  const uint32_t k0 = 0u, k1 = 42u;
  uint32_t ks[3] = {k0, k1, k0 ^ k1 ^ 0x1BD11BDAu};
  uint32_t x0 = c0 + ks[0], x1 = c1 + ks[1];
  const int rot[2][4] = {{13, 15, 26, 6}, {17, 29, 16, 24}};
#pragma unroll
  for (int i = 0; i < 5; ++i) {
#pragma unroll
    for (int j = 0; j < 4; ++j) {
      x0 += x1;
      int r = rot[i & 1][j];
      x1 = (x1 << r) | (x1 >> (32 - r));
      x1 ^= x0;
    }
    x0 += ks[(i + 1) % 3];
    x1 += ks[(i + 2) % 3] + (uint32_t)(i + 1);
  }
  o0 = x0; o1 = x1;
}

// ---------------------------------------------------------------------------
// Streaming matvec partials via WMMA f32 16x16x4; N is compile-time so every
// load in a 32-row tile is an immediate-offset load from one base pointer.
//   part[kc*N + j] = sum_{k in chunk kc} vec[k] * M[k*N + j]
// A is row-broadcast (all 16 M-rows = vec[k..k+3]); D row 0 = 16 outputs.
// Per ISA 7.12.2: A/B v0 = K0 (lanes 0-15) / K2 (lanes 16-31); v1 = K1/K3.
// ---------------------------------------------------------------------------
template <int N>
__global__ __launch_bounds__(256)
void matvec_part_kernel(const float* __restrict__ M, const float* __restrict__ vec,
                        float* __restrict__ part) {
  constexpr int NJG = N >> 4;
  int gw   = (blockIdx.x * blockDim.x + threadIdx.x) >> 5;
  int lane = threadIdx.x & 31;
  int jg   = gw % NJG;                    // power-of-two -> and/shift
  int kc   = gw / NJG;
  int jcol = (jg << 4) + (lane & 15);
  int h2   = (lane >> 4) << 1;            // 0 for lanes 0-15, 2 for lanes 16-31
  int k0   = kc * KC;
  const float* Pb = M + (size_t)(k0 + h2) * N + jcol;   // B base (row k0+h2)
  const float* Vb = vec + k0 + h2;                      // A base
  v8f acc = {};
#pragma unroll 1
  for (int kt = 0; kt < KC; kt += 32) {
    // Speculative stream-ahead (TH=0): safe unguarded, immediate offset.
    __builtin_prefetch(Pb + (size_t)64 * N, 0, 1);
#pragma unroll
    for (int kk = 0; kk < 32; kk += 4) {
      v2f a = *(const v2f*)(Vb + kk);                   // broadcast A rows
      v2f b;
      b.x = Pb[(size_t)kk * N];                         // imm-offset loads
      b.y = Pb[(size_t)(kk + 1) * N];
      acc = __builtin_amdgcn_wmma_f32_16x16x4_f32(false, a, false, b,
                                                  (short)0, acc, false, false);
    }
    Pb += (size_t)32 * N;
    Vb += 32;
  }
  if (lane < 16) part[(size_t)kc * N + jcol] = acc[0];  // D row 0
}

// Same, but computes X^T s and X^T WM0 in one pass over X (same bytes).
__global__ __launch_bounds__(256)
void matvecX_dual_kernel(const float* __restrict__ X, const float* __restrict__ s,
                         const float* __restrict__ wm0,
                         float* __restrict__ partv, float* __restrict__ partvp) {
  constexpr int N = DDIM;
  int gw   = (blockIdx.x * blockDim.x + threadIdx.x) >> 5;
  int lane = threadIdx.x & 31;
  int jg   = gw % (N >> 4);
  int kc   = gw / (N >> 4);
  int jcol = (jg << 4) + (lane & 15);
  int h2   = (lane >> 4) << 1;
  int k0   = kc * KC;
  const float* Pb = X + (size_t)(k0 + h2) * N + jcol;
  const float* Sb = s + k0 + h2;
  const float* Wb = wm0 + k0 + h2;
  v8f accv = {}, accp = {};
#pragma unroll 1
  for (int kt = 0; kt < KC; kt += 32) {
    __builtin_prefetch(Pb + (size_t)64 * N, 0, 1);
#pragma unroll
    for (int kk = 0; kk < 32; kk += 4) {
      v2f as = *(const v2f*)(Sb + kk);
      v2f aw = *(const v2f*)(Wb + kk);
      v2f b;
      b.x = Pb[(size_t)kk * N];
      b.y = Pb[(size_t)(kk + 1) * N];
      accv = __builtin_amdgcn_wmma_f32_16x16x4_f32(false, as, false, b,
                                                   (short)0, accv, false, false);
      accp = __builtin_amdgcn_wmma_f32_16x16x4_f32(false, aw, false, b,
                                                   (short)0, accp, false, false);
    }
    Pb += (size_t)32 * N;
    Sb += 32;
    Wb += 32;
  }
  if (lane < 16) {
    partv [(size_t)kc * N + jcol] = accv[0];
    partvp[(size_t)kc * N + jcol] = accp[0];
  }
}

// ---------------------------------------------------------------------------
// Gate: reduce X partials -> v, v_prev; threefry bernoulli; r = mix(s, WM0)
// Thread i handles elements i and i+4096 (one threefry counter pair).
// ---------------------------------------------------------------------------
__global__ __launch_bounds__(256)
void gate_kernel(const float* __restrict__ partv, const float* __restrict__ partvp,
                 const float* __restrict__ s, const float* __restrict__ wm0,
                 const float* __restrict__ beta_p, const float* __restrict__ bias_p,
                 float* __restrict__ r_ws, float* __restrict__ r_out) {
  int i = blockIdx.x * blockDim.x + threadIdx.x;   // 0..4095
  float v0 = 0.f, v1 = 0.f, vp0 = 0.f, vp1 = 0.f;
#pragma unroll
  for (int kc = 0; kc < KCHUNKS; ++kc) {
    const float* pv = partv  + (size_t)kc * DDIM;
    const float* pp = partvp + (size_t)kc * DDIM;
    v0  += pv[i];  v1  += pv[i + RDIM];
    vp0 += pp[i];  vp1 += pp[i + RDIM];
  }
  float beta = *beta_p, bias = *bias_p;
  uint32_t b0, b1;
  threefry2x32((uint32_t)i, (uint32_t)(i + RDIM), b0, b1);
  float u0 = __uint_as_float((b0 >> 9) | 0x3f800000u) - 1.0f;
  float u1 = __uint_as_float((b1 >> 9) | 0x3f800000u) - 1.0f;
  float ev0 = expf(beta * v0), ev1 = expf(beta * v1);
  float pr0 = (ev0 + bias) / (ev0 + bias + expf(beta * vp0));
  float pr1 = (ev1 + bias) / (ev1 + bias + expf(beta * vp1));
  float st0 = (u0 < pr0) ? 1.0f : 0.0f;
  float st1 = (u1 < pr1) ? 1.0f : 0.0f;
  float r0 = st0 * s[i]        + (1.0f - st0) * wm0[i];
  float r1 = st1 * s[i + RDIM] + (1.0f - st1) * wm0[i + RDIM];
  r_ws[i] = r0;        r_ws[i + RDIM] = r1;
  r_out[i] = r0;       r_out[i + RDIM] = r1;
}

// ---------------------------------------------------------------------------
// Finalize: p = reduce(partW); e one-hot; p_lowest = p + alpha*e[a]*r[a]*r
// (W_new never materialized: rank-1 update folded analytically.)
// ---------------------------------------------------------------------------
__global__ __launch_bounds__(256)
void finalize_kernel(const float* __restrict__ partW, const float* __restrict__ r,
                     const float* __restrict__ o, const int* __restrict__ aidx,
                     float* __restrict__ plow, float* __restrict__ e_out) {
  int i = blockIdx.x * blockDim.x + threadIdx.x;   // 0..8191
  int a = *aidx;
  float p = 0.f, pa = 0.f;
#pragma unroll
  for (int kc = 0; kc < KCHUNKS; ++kc) {
    p  += partW[(size_t)kc * DDIM + i];
    pa += partW[(size_t)kc * DDIM + a];            // redundant, L0-cached
  }
  float ev = o[a] - pa;
  e_out[i] = (i == a) ? ev : 0.0f;
  float c = ALPHA_L * ev * r[a];
  plow[i] = p + c * r[i];
}

// ---------------------------------------------------------------------------
// Fused reduce + softmax over R=4096, single 1024-thread block (deterministic)
// ---------------------------------------------------------------------------
__global__ __launch_bounds__(1024)
void softmax_kernel(const float* __restrict__ partR, float* __restrict__ probs) {
  __shared__ float sred[1024];
  int t = threadIdx.x;
  float l[4], e[4];
#pragma unroll
  for (int q = 0; q < 4; ++q) {
    int j = t + q * 1024;
    float acc = 0.f;
#pragma unroll
    for (int kc = 0; kc < KCHUNKS; ++kc) acc += partR[(size_t)kc * RDIM + j];
    l[q] = GAMMA * acc;
  }
  float m = fmaxf(fmaxf(l[0], l[1]), fmaxf(l[2], l[3]));
  sred[t] = m; __syncthreads();
  for (int sft = 512; sft > 0; sft >>= 1) {
    if (t < sft) sred[t] = fmaxf(sred[t], sred[t + sft]);
    __syncthreads();
  }
  float mg = sred[0]; __syncthreads();
  float sum = 0.f;
#pragma unroll
  for (int q = 0; q < 4; ++q) { e[q] = expf(l[q] - mg); sum += e[q]; }
  sred[t] = sum; __syncthreads();
  for (int sft = 512; sft > 0; sft >>= 1) {
    if (t < sft) sred[t] += sred[t + sft];
    __syncthreads();
  }
  float inv = 1.0f / sred[0];
#pragma unroll
  for (int q = 0; q < 4; ++q) probs[t + q * 1024] = e[q] * inv;
}

// ---------------------------------------------------------------------------
extern "C" void kernel_launch(void* const* d_in, const int* in_sizes, int n_in,
                              void* d_out, int out_size, void* d_ws, size_t ws_size,
                              hipStream_t stream) {
  (void)in_sizes; (void)n_in; (void)out_size; (void)ws_size;
  const float* s    = (const float*)d_in[0];
  const float* o    = (const float*)d_in[1];
  const float* W    = (const float*)d_in[2];
  const float* X    = (const float*)d_in[3];
  const float* Wr   = (const float*)d_in[4];
  const float* WM0  = (const float*)d_in[5];
  /* d_in[6] = elig0: eligibility trace is side state, not in outputs */
  const float* beta = (const float*)d_in[7];
  const float* bias = (const float*)d_in[8];
  const int*   aidx = (const int*)d_in[9];

  float* out   = (float*)d_out;
  float* probs = out;                 // [RDIM]
  float* e_out = out + RDIM;          // [DDIM]
  float* r_out = out + RDIM + DDIM;   // [DDIM]

  float* ws      = (float*)d_ws;
  float* partXv  = ws;                                  // 16*8192
  float* partXvp = partXv  + (size_t)KCHUNKS * DDIM;    // 16*8192
  float* partW   = partXvp + (size_t)KCHUNKS * DDIM;    // 16*8192
  float* partR   = partW   + (size_t)KCHUNKS * DDIM;    // 16*4096
  float* r_ws    = partR   + (size_t)KCHUNKS * RDIM;    // 8192
  float* plow_ws = r_ws    + DDIM;                      // 8192

  dim3 blk(256);
  const int wavesD = (DDIM / 16) * KCHUNKS;   // 8192 waves
  const int wavesR = (RDIM / 16) * KCHUNKS;   // 4096 waves

  matvecX_dual_kernel<<<wavesD / 8, blk, 0, stream>>>(X, s, WM0, partXv, partXvp);
  gate_kernel<<<RDIM / 256, blk, 0, stream>>>(partXv, partXvp, s, WM0,
                                              beta, bias, r_ws, r_out);
  matvec_part_kernel<DDIM><<<wavesD / 8, blk, 0, stream>>>(W, r_ws, partW);
  finalize_kernel<<<DDIM / 256, blk, 0, stream>>>(partW, r_ws, o, aidx,
                                                  plow_ws, e_out);
  matvec_part_kernel<RDIM><<<wavesR / 8, blk, 0, stream>>>(Wr, plow_ws, partR);
  softmax_kernel<<<1, 1024, 0, stream>>>(partR, probs);
}